// StructEncoder_38371237822905
// MI455X (gfx1250) — compile-verified
//
#include <hip/hip_runtime.h>
#include <hip/hip_bf16.h>
#include <math.h>

#define EPS 1e-8f

typedef __attribute__((ext_vector_type(2)))  float v2f;
typedef __attribute__((ext_vector_type(8)))  float v8f;

// ---------------------------------------------------------------- utilities
__global__ void fill_f32(float* __restrict__ p, float v, int n) {
    int i = blockIdx.x * blockDim.x + threadIdx.x;
    if (i < n) p[i] = v;
}

// ------------------------------------------------- weighted degree (scatter)
__global__ void deg_scatter(const int* __restrict__ src,
                            const float* __restrict__ w,
                            float* __restrict__ deg, int E) {
    int stride = gridDim.x * blockDim.x;
    for (int e = blockIdx.x * blockDim.x + threadIdx.x; e < E; e += stride)
        atomicAdd(deg + src[e], w[e]);
}

// -------------------------------------------- pagerank message pass (scatter)
__global__ void pr_scatter(const int* __restrict__ src,
                           const int* __restrict__ dst,
                           const float* __restrict__ w,
                           const float* __restrict__ pr,
                           float* __restrict__ pr_next, int E) {
    int stride = gridDim.x * blockDim.x;
    for (int e = blockIdx.x * blockDim.x + threadIdx.x; e < E; e += stride)
        atomicAdd(pr_next + dst[e], pr[src[e]] * w[e]);
}

// ----------------------------------------------------------- sum reduction
__global__ void reduce_sum(const float* __restrict__ v,
                           float* __restrict__ out_slot, int n) {
    __shared__ float sm[256];
    int tid = threadIdx.x;
    int i = blockIdx.x * 256 + tid;
    sm[tid] = (i < n) ? v[i] : 0.0f;
    __syncthreads();
    for (int s = 128; s > 0; s >>= 1) {
        if (tid < s) sm[tid] += sm[tid + s];
        __syncthreads();
    }
    if (tid == 0) atomicAdd(out_slot, sm[0]);
}

// ----------------------------------------------------------- pr normalize
__global__ void pr_norm(const float* __restrict__ pr_next,
                        float* __restrict__ pr,
                        const float* __restrict__ sum_slot, int n) {
    int i = blockIdx.x * blockDim.x + threadIdx.x;
    if (i < n) {
        float s = fmaxf(*sum_slot, EPS);
        pr[i] = pr_next[i] / s;
    }
}

// ------------------------------ column stats: sum & sumsq for [deg,logdeg,pr]
// accum[4..6] = sums, accum[7..9] = sums of squares
__global__ void stats(const float* __restrict__ deg,
                      const float* __restrict__ pr,
                      float* __restrict__ accum, int n) {
    __shared__ float sm[256];
    int tid = threadIdx.x;
    int i = blockIdx.x * 256 + tid;
    float d = 0.f, l = 0.f, p = 0.f;
    if (i < n) {
        d = fmaxf(deg[i], EPS);
        l = logf(d);
        p = pr[i];
    }
    float vals[6] = {d, l, p, d * d, l * l, p * p};
    #pragma unroll
    for (int k = 0; k < 6; ++k) {
        sm[tid] = vals[k];
        __syncthreads();
        for (int s = 128; s > 0; s >>= 1) {
            if (tid < s) sm[tid] += sm[tid + s];
            __syncthreads();
        }
        if (tid == 0) atomicAdd(&accum[4 + k], sm[0]);
        __syncthreads();
    }
}

// ------------- finalize: accum[10..12]=mean, accum[13..15]=1/std (ddof=1)
__global__ void finalize_stats(float* __restrict__ accum, int n) {
    int k = threadIdx.x;
    if (k < 3) {
        float su = accum[4 + k], ss = accum[7 + k];
        float m  = su / (float)n;
        float var = (ss - su * m) / (float)(n - 1);
        float sd = fmaxf(sqrtf(fmaxf(var, 0.0f)), EPS);
        accum[10 + k] = m;
        accum[13 + k] = 1.0f / sd;
    }
}

// --------------------------------------------------------------------------
// Per-lane A operand for V_WMMA_F32_16X16X4_F32 (16x4 f32 A-matrix layout):
//   lanes 0-15 : {K=0,K=1} = {s0,s1}   lanes 16-31 : {K=2,K=3} = {s2,0}
// --------------------------------------------------------------------------
__device__ __forceinline__ v2f make_a(const float* __restrict__ deg,
                                      const float* __restrict__ pr,
                                      const float* __restrict__ accum,
                                      int row, int half, int N) {
    const float m0 = accum[10], m1 = accum[11], m2 = accum[12];
    const float i0 = accum[13], i1 = accum[14], i2 = accum[15];
    const int r = (row < N) ? row : 0;
    float d0 = fmaxf(deg[r], EPS);
    float s0 = (d0 - m0) * i0;
    float s1 = (logf(d0) - m1) * i1;
    float s2 = (pr[r] - m2) * i2;
    if (row >= N) { s0 = s1 = s2 = 0.0f; }
    v2f a;
    a.x = (half == 0) ? s0 : s2;
    a.y = (half == 0) ? s1 : 0.0f;
    return a;
}

// --------------------------------------------------------------------------
// out[16 rows x DIM cols] per wave via V_WMMA_F32_16X16X4_F32.
//   B (4x16 f32):  lanes 0-15 {K=0,K=1} = {W0[c],W1[c]}; lanes 16-31 {W2[c],0}
//   C/D (16x16):   VGPR g -> M = g (lanes 0-15) / g+8 (lanes 16-31), N = lane%16
// Bias folded into C: D = s@W + b in one instruction per 16x16 tile.
// Uniform fulltile branch hoisted OUTSIDE the j loop: hot path is
// load-B -> wmma -> clause of 8 unguarded b32 stores.
// --------------------------------------------------------------------------
template <int DIMC>
__global__ void __launch_bounds__(32)
out_wmma(const float* __restrict__ deg, const float* __restrict__ pr,
         const float* __restrict__ accum, const float* __restrict__ W,
         const float* __restrict__ bias, float* __restrict__ out, int N) {
    const int lane = threadIdx.x;
    const int half = lane >> 4;
    const int l16  = lane & 15;
    const int tile = blockIdx.x;

    const int row = tile * 16 + l16;
    const v2f a = make_a(deg, pr, accum, row, half, N);
    const int mbase = tile * 16 + half * 8;

    if (tile * 16 + 16 <= N) {               // uniform: full tile (hot path)
        #pragma unroll
        for (int j = 0; j < DIMC / 16; ++j) {
            const int col = j * 16 + l16;
            v2f bm;
            bm.x = (half == 0) ? W[col]        : W[2 * DIMC + col];
            bm.y = (half == 0) ? W[DIMC + col] : 0.0f;
            const float bb = bias[col];
            v8f c;
            #pragma unroll
            for (int g = 0; g < 8; ++g) c[g] = bb;
            v8f dmat = __builtin_amdgcn_wmma_f32_16x16x4_f32(
                false, a, false, bm, (short)0, c, false, false);
            float* outp = out + (size_t)mbase * DIMC + col;
            #pragma unroll
            for (int g = 0; g < 8; ++g) outp[(size_t)g * DIMC] = dmat[g];
        }
    } else {                                  // uniform: ragged tail (cold)
        #pragma unroll
        for (int j = 0; j < DIMC / 16; ++j) {
            const int col = j * 16 + l16;
            v2f bm;
            bm.x = (half == 0) ? W[col]        : W[2 * DIMC + col];
            bm.y = (half == 0) ? W[DIMC + col] : 0.0f;
            const float bb = bias[col];
            v8f c;
            #pragma unroll
            for (int g = 0; g < 8; ++g) c[g] = bb;
            v8f dmat = __builtin_amdgcn_wmma_f32_16x16x4_f32(
                false, a, false, bm, (short)0, c, false, false);
            float* outp = out + (size_t)mbase * DIMC + col;
            #pragma unroll
            for (int g = 0; g < 8; ++g)
                if (mbase + g < N) outp[(size_t)g * DIMC] = dmat[g];
        }
    }
}

// Runtime-DIM fallback (same structure, runtime loop bound).
__global__ void __launch_bounds__(32)
out_wmma_dyn(const float* __restrict__ deg, const float* __restrict__ pr,
             const float* __restrict__ accum, const float* __restrict__ W,
             const float* __restrict__ bias, float* __restrict__ out,
             int N, int DIM) {
    const int lane = threadIdx.x;
    const int half = lane >> 4;
    const int l16  = lane & 15;
    const int tile = blockIdx.x;

    const int row = tile * 16 + l16;
    const v2f a = make_a(deg, pr, accum, row, half, N);
    const int mbase = tile * 16 + half * 8;

    if (tile * 16 + 16 <= N) {
        for (int j = 0; j < DIM / 16; ++j) {
            const int col = j * 16 + l16;
            v2f bm;
            bm.x = (half == 0) ? W[col]       : W[2 * DIM + col];
            bm.y = (half == 0) ? W[DIM + col] : 0.0f;
            const float bb = bias[col];
            v8f c;
            #pragma unroll
            for (int g = 0; g < 8; ++g) c[g] = bb;
            v8f dmat = __builtin_amdgcn_wmma_f32_16x16x4_f32(
                false, a, false, bm, (short)0, c, false, false);
            float* outp = out + (size_t)mbase * DIM + col;
            #pragma unroll
            for (int g = 0; g < 8; ++g) outp[(size_t)g * DIM] = dmat[g];
        }
    } else {
        for (int j = 0; j < DIM / 16; ++j) {
            const int col = j * 16 + l16;
            v2f bm;
            bm.x = (half == 0) ? W[col]       : W[2 * DIM + col];
            bm.y = (half == 0) ? W[DIM + col] : 0.0f;
            const float bb = bias[col];
            v8f c;
            #pragma unroll
            for (int g = 0; g < 8; ++g) c[g] = bb;
            v8f dmat = __builtin_amdgcn_wmma_f32_16x16x4_f32(
                false, a, false, bm, (short)0, c, false, false);
            float* outp = out + (size_t)mbase * DIM + col;
            #pragma unroll
            for (int g = 0; g < 8; ++g)
                if (mbase + g < N) outp[(size_t)g * DIM] = dmat[g];
        }
    }
}

// --------------------------------------------------------------------------
extern "C" void kernel_launch(void* const* d_in, const int* in_sizes, int n_in,
                              void* d_out, int out_size, void* d_ws, size_t ws_size,
                              hipStream_t stream) {
    (void)n_in; (void)out_size; (void)ws_size;
    // inputs: 0=x [N,DIM] f32 (shape only), 1=edge_index [2,E] int,
    //         2=edge_weight_norm [E] f32, 3=W [3,DIM] f32, 4=b [DIM] f32
    const int*   ei = (const int*)d_in[1];
    const float* w  = (const float*)d_in[2];
    const float* Wm = (const float*)d_in[3];
    const float* bv = (const float*)d_in[4];
    float* out = (float*)d_out;

    const int DIM = in_sizes[4];
    const int N   = in_sizes[0] / DIM;
    const int E   = in_sizes[2];
    const int* src = ei;
    const int* dst = ei + E;

    // workspace: deg[N] | pr[N] | pr_next[N] | accum[16]
    float* deg   = (float*)d_ws;
    float* pr    = deg + N;
    float* prn   = pr + N;
    float* accum = prn + N;   // [0..2]=pr sums, [4..9]=stats, [10..15]=mean/invstd

    const int tb = 256;
    const int gN = (N + tb - 1) / tb;
    int gE = (E + tb * 4 - 1) / (tb * 4);      // ~4 edges/thread, grid-stride
    if (gE < 1) gE = 1;

    fill_f32<<<gN, tb, 0, stream>>>(deg, 0.0f, N);
    fill_f32<<<gN, tb, 0, stream>>>(pr, 1.0f / (float)N, N);
    fill_f32<<<1, 32, 0, stream>>>(accum, 0.0f, 16);

    deg_scatter<<<gE, tb, 0, stream>>>(src, w, deg, E);

    for (int t = 0; t < 3; ++t) {
        fill_f32<<<gN, tb, 0, stream>>>(prn, 0.0f, N);
        pr_scatter<<<gE, tb, 0, stream>>>(src, dst, w, pr, prn, E);
        reduce_sum<<<gN, tb, 0, stream>>>(prn, accum + t, N);
        pr_norm<<<gN, tb, 0, stream>>>(prn, pr, accum + t, N);
    }

    stats<<<gN, tb, 0, stream>>>(deg, pr, accum, N);
    finalize_stats<<<1, 32, 0, stream>>>(accum, N);

    const int tiles = (N + 15) / 16;
    if (DIM == 256)
        out_wmma<256><<<tiles, 32, 0, stream>>>(deg, pr, accum, Wm, bv, out, N);
    else
        out_wmma_dyn<<<tiles, 32, 0, stream>>>(deg, pr, accum, Wm, bv, out, N, DIM);
}